// SqueezeExciteBlock_90417651516343
// MI455X (gfx1250) — compile-verified
//
#include <hip/hip_runtime.h>

typedef float v2f __attribute__((ext_vector_type(2)));
typedef float v4f __attribute__((ext_vector_type(4)));
typedef float v8f __attribute__((ext_vector_type(8)));

#define SE_B 16
#define SE_C 512
#define SE_L 8192
#define SE_H 32   // C / 16

// ---------------------------------------------------------------------------
// Kernel 1: per-(b,c) row mean over L=8192.  One block (256 thr) per row.
// 8192 blocks, fully coalesced float4 loads; shuffle + LDS reduction.
// ---------------------------------------------------------------------------
__global__ void se_mean_kernel(const float* __restrict__ x, float* __restrict__ s) {
    const int row = blockIdx.x;                       // [0, B*C)
    const v4f* xr = (const v4f*)(x + (size_t)row * SE_L);
    const int t = threadIdx.x;

    float sum = 0.0f;
#pragma unroll
    for (int k = 0; k < 8; ++k) {                     // 2048 float4 per row
        v4f v = xr[t + k * 256];
        sum += v.x + v.y + v.z + v.w;
    }
    // wave32 reduction
#pragma unroll
    for (int off = 16; off > 0; off >>= 1)
        sum += __shfl_down(sum, off, 32);

    __shared__ float wsum[8];
    if ((t & 31) == 0) wsum[t >> 5] = sum;
    __syncthreads();
    if (t == 0) {
        float tot = 0.0f;
#pragma unroll
        for (int w = 0; w < 8; ++w) tot += wsum[w];
        s[row] = tot * (1.0f / (float)SE_L);
    }
}

// ---------------------------------------------------------------------------
// Kernel 2: tiny MLP with V_WMMA_F32_16X16X4_F32 (fp32 WMMA).
//   GEMM1: h[16,32]  = relu(s[16,512] @ w1^T + b1)   (wave 0, 2 N-tiles)
//   GEMM2: g[16,512] = sigmoid(h @ w2^T + b2)        (4 waves x 8 N-tiles)
// Layouts per ISA 7.12.2 (f32):
//   A 16x4 : lane -> m = lane&15 ; K = vgpr + 2*(lane>>4)
//   B 4x16 : lane -> n = lane&15 ; K = vgpr + 2*(lane>>4)
//   C/D    : lane,vgpr v -> D[m = v + 8*(lane>>4)][n = lane&15]
// ---------------------------------------------------------------------------
__global__ void se_mlp_kernel(const float* __restrict__ s,
                              const float* __restrict__ w1,
                              const float* __restrict__ b1,
                              const float* __restrict__ w2,
                              const float* __restrict__ b2,
                              float* __restrict__ g) {
    __shared__ float hlds[SE_B * SE_H];               // 16 x 32 = 2 KB

    const int tid  = threadIdx.x;
    const int lane = tid & 31;
    const int wave = tid >> 5;
    const int lh   = lane >> 4;                       // lane half (0/1)
    const int ln   = lane & 15;                       // 0..15

    if (wave == 0) {
        // ---- GEMM1: M=16(b) N=32(j) K=512(c) ----
        v8f acc0 = {};                                // j tile 0..15
        v8f acc1 = {};                                // j tile 16..31
        for (int k0 = 0; k0 < SE_C; k0 += 4) {
            const int k = k0 + 2 * lh;
            v2f a;  a.x  = s[ln * SE_C + k];          a.y  = s[ln * SE_C + k + 1];
            v2f b0; b0.x = w1[ln * SE_C + k];         b0.y = w1[ln * SE_C + k + 1];
            v2f b1v; b1v.x = w1[(ln + 16) * SE_C + k]; b1v.y = w1[(ln + 16) * SE_C + k + 1];
            acc0 = __builtin_amdgcn_wmma_f32_16x16x4_f32(false, a, false, b0,
                                                         (short)0, acc0, false, false);
            acc1 = __builtin_amdgcn_wmma_f32_16x16x4_f32(false, a, false, b1v,
                                                         (short)0, acc1, false, false);
        }
#pragma unroll
        for (int v = 0; v < 8; ++v) {
            const int m = v + 8 * lh;                 // batch row
            float h0 = acc0[v] + b1[ln];
            float h1 = acc1[v] + b1[ln + 16];
            hlds[m * SE_H + ln]      = fmaxf(h0, 0.0f);
            hlds[m * SE_H + ln + 16] = fmaxf(h1, 0.0f);
        }
    }
    __syncthreads();

    // ---- GEMM2: M=16(b) N=512(c) K=32(j); wave w owns c-tiles [w*8, w*8+8) ----
    for (int t = 0; t < 8; ++t) {
        const int cbase = (wave * 8 + t) * 16;
        v8f acc = {};
#pragma unroll
        for (int k0 = 0; k0 < SE_H; k0 += 4) {
            const int k = k0 + 2 * lh;
            v2f a; a.x = hlds[ln * SE_H + k];             a.y = hlds[ln * SE_H + k + 1];
            v2f b; b.x = w2[(cbase + ln) * SE_H + k];     b.y = w2[(cbase + ln) * SE_H + k + 1];
            acc = __builtin_amdgcn_wmma_f32_16x16x4_f32(false, a, false, b,
                                                        (short)0, acc, false, false);
        }
        const int c = cbase + ln;
        const float bias = b2[c];
#pragma unroll
        for (int v = 0; v < 8; ++v) {
            const int m = v + 8 * lh;
            float p = acc[v] + bias;
            g[m * SE_C + c] = 1.0f / (1.0f + __expf(-p));
        }
    }
}

// ---------------------------------------------------------------------------
// Kernel 3: out = x * g[b,c].  One block per row; non-temporal stores so the
// streamed output doesn't evict x lines that kernel 1 left in L2.
// ---------------------------------------------------------------------------
__global__ void se_scale_kernel(const float* __restrict__ x,
                                const float* __restrict__ g,
                                float* __restrict__ out) {
    const int row = blockIdx.x;
    const float gv = g[row];
    const v4f* xr = (const v4f*)(x + (size_t)row * SE_L);
    v4f* orow     = (v4f*)(out + (size_t)row * SE_L);
    const int t = threadIdx.x;

#pragma unroll
    for (int k = 0; k < 8; ++k) {
        v4f v = xr[t + k * 256];
        v *= gv;
        __builtin_nontemporal_store(v, &orow[t + k * 256]);
    }
}

// ---------------------------------------------------------------------------
extern "C" void kernel_launch(void* const* d_in, const int* in_sizes, int n_in,
                              void* d_out, int out_size, void* d_ws, size_t ws_size,
                              hipStream_t stream) {
    const float* x  = (const float*)d_in[0];  // [16,512,8192]
    const float* w1 = (const float*)d_in[1];  // [32,512]
    const float* b1 = (const float*)d_in[2];  // [32]
    const float* w2 = (const float*)d_in[3];  // [512,32]
    const float* b2 = (const float*)d_in[4];  // [512]
    float* out = (float*)d_out;

    float* s = (float*)d_ws;                  // [16*512] means
    float* g = s + SE_B * SE_C;               // [16*512] gates

    const int rows = SE_B * SE_C;             // 8192
    se_mean_kernel<<<rows, 256, 0, stream>>>(x, s);
    se_mlp_kernel<<<1, 128, 0, stream>>>(s, w1, b1, w2, b2, g);
    se_scale_kernel<<<rows, 256, 0, stream>>>(x, g, out);
}